// HashEmbedder_9689446220224
// MI455X (gfx1250) — compile-verified
//
#include <hip/hip_runtime.h>
#include <hip/hip_bf16.h>
#include <math.h>

// ---------------- problem constants (match reference) ----------------
#define N_LEVELS   16
#define F_LEVEL    2
#define LOG2_T     19
#define TSIZE      (1u << LOG2_T)
#define TMASK      (TSIZE - 1u)
#define PRIME1     2654435761u
#define PRIME2     805459861u

#define PTS_PER_BLOCK 16
#define BLOCK_THREADS (PTS_PER_BLOCK * N_LEVELS)   // 256 = 8 waves (wave32)

typedef float v2f  __attribute__((ext_vector_type(2)));
typedef unsigned int u32x4 __attribute__((ext_vector_type(4)));
typedef int i32x4 __attribute__((ext_vector_type(4)));
typedef int i32x8 __attribute__((ext_vector_type(8)));

#if __has_builtin(__builtin_amdgcn_tensor_load_to_lds) && __has_builtin(__builtin_amdgcn_s_wait_tensorcnt)
#define USE_TDM 1
#endif

__device__ __forceinline__ float fast_exp2(float v) {
#if __has_builtin(__builtin_amdgcn_exp2f)
    return __builtin_amdgcn_exp2f(v);   // single v_exp_f32 (arg range here is [0,5], no denorm issues)
#else
    return exp2f(v);
#endif
}

__global__ __launch_bounds__(BLOCK_THREADS) void hashgrid_fwd_kernel(
    const float* __restrict__ x,        // [B,3]
    const float* __restrict__ emb,      // [L, 2^19, 2]
    const float* __restrict__ box_min,  // [3]
    const float* __restrict__ box_max,  // [3]
    float* __restrict__ out,            // [B, 32]
    int npts)
{
    __shared__ float sx[PTS_PER_BLOCK * 3];

    const int tid = threadIdx.x;
    const int blk = blockIdx.x;
    const long long base_elem = (long long)blk * (PTS_PER_BLOCK * 3);
    const int n_elem = (int)min((long long)(PTS_PER_BLOCK * 3),
                                (long long)npts * 3 - base_elem);

    // ---- stage this block's 16 points (48 floats) into LDS via TDM (async DMA) ----
#ifdef USE_TDM
    if (tid == 0 && n_elem > 0) {
        unsigned long long gaddr = (unsigned long long)(const void*)(x + base_elem);
        unsigned lds_addr = (unsigned)(size_t)(&sx[0]);   // LDS byte offset (low 32 bits of flat)
        u32x4 g0;
        g0.x = 1u;                                   // count=1 valid descriptor, user mode
        g0.y = lds_addr;                             // lds_addr
        g0.z = (unsigned)(gaddr & 0xffffffffull);    // global_addr[31:0]
        g0.w = (unsigned)((gaddr >> 32) & 0x1ffffffu) | (2u << 30); // addr[56:32] | type=2
        i32x8 g1;
        g1[0] = 0x00020000;                          // data_size=2 (4 bytes), no multicast
        g1[1] = (n_elem << 16);                      // tensor_dim0[15:0] = n_elem
        g1[2] = (1 << 16);                           // tensor_dim0 hi=0, tensor_dim1=1
        g1[3] = (n_elem << 16);                      // tile_dim0 = n_elem (1D tile)
        g1[4] = 0;                                   // tile_dim1=0 (unused), tile_dim2=0
        g1[5] = n_elem;                              // tensor_dim0_stride
        g1[6] = 0;
        g1[7] = 0;
        i32x4 z4 = {0, 0, 0, 0};                     // groups 2/3 unused (<=2D)
        i32x8 z8 = {0, 0, 0, 0, 0, 0, 0, 0};
        __builtin_amdgcn_tensor_load_to_lds(g0, g1, z4, z4, z8, 0);
        __builtin_amdgcn_s_wait_tensorcnt(0);
    }
    __syncthreads();
#else
    if (tid < PTS_PER_BLOCK * 3 && tid < n_elem)
        sx[tid] = x[base_elem + tid];
    __syncthreads();
#endif

    const int level = tid & (N_LEVELS - 1);
    const int lp    = tid >> 4;                      // local point index
    const long long pt = (long long)blk * PTS_PER_BLOCK + lp;
    if (pt >= npts) return;

    // ---- uniform box (scalar loads) ----
    const float bn0 = box_min[0], bn1 = box_min[1], bn2 = box_min[2];
    const float bx0 = box_max[0], bx1 = box_max[1], bx2 = box_max[2];

    const float px = sx[lp * 3 + 0];
    const float py = sx[lp * 3 + 1];
    const float pz = sx[lp * 3 + 2];

    // per-level resolution (f32, matches reference values):
    // b = exp((ln512-ln16)/15); res = floor(16 * b^level), b^l via exp2(l*log2(b))
    // (bgrow and lb constant-fold at compile time; one v_exp_f32 at runtime)
    const float bgrow = expf((logf(512.0f) - logf(16.0f)) * (1.0f / 15.0f));
    const float lb    = log2f(bgrow);
    const float resf  = floorf(16.0f * fast_exp2((float)level * lb));

    const float gx = (bx0 - bn0) / resf;
    const float gy = (bx1 - bn1) / resf;
    const float gz = (bx2 - bn2) / resf;

    const float flx = floorf((px - bn0) / gx);
    const float fly = floorf((py - bn1) / gy);
    const float flz = floorf((pz - bn2) / gz);

    const float wx = (px - (flx * gx + bn0)) / gx;
    const float wy = (py - (fly * gy + bn1)) / gy;
    const float wz = (pz - (flz * gz + bn2)) / gz;
    const float ox = 1.0f - wx, oy = 1.0f - wy, oz = 1.0f - wz;

    // hash pre-products (uint32 wraparound, primes {1, 2654435761, 805459861})
    const unsigned hx0 = (unsigned)(int)flx;
    const unsigned hy0 = (unsigned)(int)fly * PRIME1;
    const unsigned hz0 = (unsigned)(int)flz * PRIME2;
    const unsigned hx1 = hx0 + 1u;
    const unsigned hy1 = hy0 + PRIME1;
    const unsigned hz1 = hz0 + PRIME2;

    const v2f* __restrict__ tab = (const v2f*)emb + (size_t)level * TSIZE;

    // 8 corner gathers: compute all indices/weights, then issue all loads (ILP)
    unsigned h[8];
    float cw[8];
#pragma unroll
    for (int c = 0; c < 8; ++c) {
        const int i = (c >> 2) & 1, j = (c >> 1) & 1, k = c & 1;
        h[c]  = ((i ? hx1 : hx0) ^ (j ? hy1 : hy0) ^ (k ? hz1 : hz0)) & TMASK;
        cw[c] = (i ? wx : ox) * (j ? wy : oy) * (k ? wz : oz);
    }
    v2f f[8];
#pragma unroll
    for (int c = 0; c < 8; ++c) f[c] = tab[h[c]];    // 8 outstanding global_load_b64

    float s0 = 0.0f, s1 = 0.0f;
#pragma unroll
    for (int c = 0; c < 8; ++c) {
        s0 = fmaf(cw[c], f[c].x, s0);
        s1 = fmaf(cw[c], f[c].y, s1);
    }

    // streaming (non-temporal) store: keep the 64MB tables resident in the 192MB L2
    v2f o; o.x = s0; o.y = s1;
    __builtin_nontemporal_store(
        o, (v2f*)(out + (size_t)pt * (N_LEVELS * F_LEVEL) + level * F_LEVEL));
}

extern "C" void kernel_launch(void* const* d_in, const int* in_sizes, int n_in,
                              void* d_out, int out_size, void* d_ws, size_t ws_size,
                              hipStream_t stream) {
    const float* x       = (const float*)d_in[0];
    const float* emb     = (const float*)d_in[1];
    const float* box_min = (const float*)d_in[2];
    const float* box_max = (const float*)d_in[3];
    float* out           = (float*)d_out;

    const int npts = in_sizes[0] / 3;
    const int nblocks = (npts + PTS_PER_BLOCK - 1) / PTS_PER_BLOCK;

    hashgrid_fwd_kernel<<<nblocks, BLOCK_THREADS, 0, stream>>>(
        x, emb, box_min, box_max, out, npts);
}